// GraphAttention_60687887892712
// MI455X (gfx1250) — compile-verified
//
#include <hip/hip_runtime.h>

typedef __attribute__((ext_vector_type(2))) float v2f;
typedef __attribute__((ext_vector_type(4))) float v4f;
typedef __attribute__((ext_vector_type(8))) float v8f;

// Monotonic float -> uint key so unsigned atomicMax implements float segment-max.
__device__ __forceinline__ unsigned fkey(float f) {
    unsigned u = __float_as_uint(f);
    return (u & 0x80000000u) ? ~u : (u | 0x80000000u);
}
__device__ __forceinline__ float funkey(unsigned k) {
    unsigned u = (k & 0x80000000u) ? (k & 0x7fffffffu) : ~k;
    return __uint_as_float(u);
}

// ---------------------------------------------------------------------------
// Kernel 0: zero the output (N*64 f32) and the smax/denom region (N*8 words).
// ---------------------------------------------------------------------------
__global__ void gat_init(float* __restrict__ out, float* __restrict__ mw,
                         long nout, long nmw) {
    long i = (long)blockIdx.x * blockDim.x + threadIdx.x;
    if (i < nout) out[i] = 0.0f;
    if (i < nmw)  mw[i]  = 0.0f;   // key 0 < key(-inf)=0x007FFFFF, valid -infty
}

// ---------------------------------------------------------------------------
// Kernel 1: per-edge per-head QK dots via V_WMMA_F32_16X16X4_F32.
// One wave handles 16 edges. For head h: D = Q(16x16) x K^T(16x16) as four
// chained K=4 WMMAs; diag(D) = the 16 per-edge dot products (exact f32).
// A layout (16x4): lane L holds M=L&15, K={2*(L>>4), 2*(L>>4)+1}.
// B layout (4x16): lane L holds N=L&15, K={2*(L>>4), 2*(L>>4)+1}.
// => both operands: lane L loads row e0+(L&15), cols d0..d0+1, d0=cb+2*(L>>4).
// ---------------------------------------------------------------------------
__global__ void gat_scores(const float* __restrict__ q, const float* __restrict__ k,
                           const int* __restrict__ dst,
                           float* __restrict__ scores, unsigned* __restrict__ smax,
                           int E) {
    const int lane = threadIdx.x & 31;
    const int wave = threadIdx.x >> 5;
    const long e0  = ((long)blockIdx.x * 8 + wave) * 16;
    const int  m   = lane & 15;
    const int  sel = lane >> 4;

    long e     = e0 + m;
    long eload = (e < E) ? e : (long)(E - 1);     // clamp: keep EXEC all-ones for WMMA
    const float* qrow = q + eload * 64;
    const float* krow = k + eload * 64;

    const int s3  = lane & 7;
    const int src = (m & 8) ? (m + 16) : m;       // diag (m,m) lives on this lane

#pragma unroll
    for (int h = 0; h < 4; ++h) {
        v8f c = {};
#pragma unroll
        for (int cc = 0; cc < 4; ++cc) {
            const int d0 = h * 16 + cc * 4 + 2 * sel;
            // stream-once data: non-temporal loads keep L2 free for atomics
            v2f a = __builtin_nontemporal_load((const v2f*)(qrow + d0));
            v2f b = __builtin_nontemporal_load((const v2f*)(krow + d0));
            c = __builtin_amdgcn_wmma_f32_16x16x4_f32(
                    false, a, false, b, (short)0, c, false, false);
        }
        // Branchless binary mux tree: lane L selects accumulator VGPR (L&7).
        float t0 = (s3 & 1) ? c[1] : c[0];
        float t1 = (s3 & 1) ? c[3] : c[2];
        float t2 = (s3 & 1) ? c[5] : c[4];
        float t3 = (s3 & 1) ? c[7] : c[6];
        float u0 = (s3 & 2) ? t1 : t0;
        float u1 = (s3 & 2) ? t3 : t2;
        float dv = (s3 & 4) ? u1 : u0;
        float s = __shfl(dv, src, 32);            // lanes 0..15 now hold diag(D)
        s *= 0.125f;                              // TEMP = 64^-0.5
        s = fmaxf(s, 0.2f * s);                   // LeakyReLU(0.2), branchless
        if (lane < 16 && e < E) {
            __builtin_nontemporal_store(s, scores + e * 4 + h);
            int n = dst[e];
            atomicMax(&smax[(long)n * 4 + h], fkey(s));
        }
    }
}

// ---------------------------------------------------------------------------
// Kernel 2: ex = exp(s - smax[dst]); overwrite scratch; segment-sum denom.
// ---------------------------------------------------------------------------
__global__ void gat_softmax(const int* __restrict__ dst, float* __restrict__ scores,
                            const unsigned* __restrict__ smax, float* __restrict__ denom,
                            long total) {
    long i = (long)blockIdx.x * blockDim.x + threadIdx.x;
    if (i >= total) return;
    long e = i >> 2;
    int  h = (int)(i & 3);
    int  n = dst[e];
    float s  = __builtin_nontemporal_load(scores + i);
    float mx = funkey(smax[(long)n * 4 + h]);
    float ex = expf(s - mx);
    __builtin_nontemporal_store(ex, scores + i);
    unsafeAtomicAdd(&denom[(long)n * 4 + h], ex);
}

// ---------------------------------------------------------------------------
// Kernel 3: weights = (ex/denom + bias) * mask; scatter w*V into out.
// 16 lanes per edge; each lane handles 4 hidden dims (float4, coalesced).
// Out is N*64*4B = 12.8 MB -> atomics land in the 192 MB L2.
// ---------------------------------------------------------------------------
__global__ void gat_scatter(const float* __restrict__ values, const float* __restrict__ bias,
                            const float* __restrict__ mask, const int* __restrict__ dst,
                            const float* __restrict__ ex, const float* __restrict__ denom,
                            float* __restrict__ out, int E) {
    long t = (long)blockIdx.x * blockDim.x + threadIdx.x;
    long e = t >> 4;
    if (e >= E) return;
    int g = (int)(t & 15);
    int h = g >> 2;
    int n = dst[e];
    float w = __builtin_nontemporal_load(ex + e * 4 + h) / denom[(long)n * 4 + h];
    w = (w + __builtin_nontemporal_load(bias + e * 4 + h)) *
        __builtin_nontemporal_load(mask + e);
    v4f v = __builtin_nontemporal_load((const v4f*)(values + e * 64 + g * 4));
    float* o = out + (long)n * 64 + g * 4;
    unsafeAtomicAdd(o + 0, w * v.x);
    unsafeAtomicAdd(o + 1, w * v.y);
    unsafeAtomicAdd(o + 2, w * v.z);
    unsafeAtomicAdd(o + 3, w * v.w);
}

// ---------------------------------------------------------------------------
extern "C" void kernel_launch(void* const* d_in, const int* in_sizes, int n_in,
                              void* d_out, int out_size, void* d_ws, size_t ws_size,
                              hipStream_t stream) {
    const float* keys    = (const float*)d_in[0];
    const float* queries = (const float*)d_in[1];
    const float* values  = (const float*)d_in[2];
    const float* bias    = (const float*)d_in[3];
    const float* mask    = (const float*)d_in[4];
    const int*   dst     = (const int*)d_in[5];

    const int E = in_sizes[0] / 64;   // keys is (E, 64)
    const int N = out_size / 64;      // output is (N, 64); num_nodes lives on device
    float* out = (float*)d_out;

    // Workspace layout: scores/ex (E*4 f32) | smax (N*4 u32) | denom (N*4 f32)
    float*    scores = (float*)d_ws;
    unsigned* smax   = (unsigned*)(scores + (size_t)E * 4);
    float*    denom  = (float*)(smax + (size_t)N * 4);

    const long nout = (long)N * 64;
    const long nmw  = (long)N * 8;    // smax + denom are contiguous
    gat_init<<<(unsigned)((nout + 255) / 256), 256, 0, stream>>>(out, (float*)smax, nout, nmw);

    // 8 waves/block, 16 edges/wave -> 128 edges/block
    gat_scores<<<(unsigned)((E + 127) / 128), 256, 0, stream>>>(
        queries, keys, dst, scores, smax, E);

    const long tot = (long)E * 4;
    gat_softmax<<<(unsigned)((tot + 255) / 256), 256, 0, stream>>>(
        dst, scores, smax, denom, tot);

    gat_scatter<<<(unsigned)(((long)E * 16 + 255) / 256), 256, 0, stream>>>(
        values, bias, mask, dst, scores, denom, out, E);
}